// LearnableDictionaryEncoding_57569741636285
// MI455X (gfx1250) — compile-verified
//
#include <hip/hip_runtime.h>
#include <math.h>
#include <stdint.h>

// CDNA5 / gfx1250. wave32. WMMA f32 16x16x4 (full-precision matrix path),
// async global->LDS staging (ASYNCcnt), global_prefetch for the dictionary.

typedef float v2f __attribute__((ext_vector_type(2)));
typedef float v8f __attribute__((ext_vector_type(8)));

#define Lsz   1024
#define Bsz   8
#define Dsz   256
#define Ksz   64
#define ROWS  32            // L-rows per block
#define CHNK  (Lsz / ROWS)  // 32 chunks per batch
#define XPAD  4
#define XSTR  (Dsz + XPAD)  // 260: bank stride 4 mod 64 -> conflict-free column reads
#define WSTR  (Ksz + 1)     // 65

// workspace layout (floats):
//   [0,64)                      d2[k]   = ||Dict[k]||^2
//   [64,128)                    nw[k]   = softmax(cluster_weights)
//   [128, 128+B*CHNK*Dsz)       S1p     per-chunk partial weighted row sums
//   [...,  +B*CHNK*Ksz)         Cp      per-chunk partial column sums of W

__global__ __launch_bounds__(64)
void lde_prep(const float* __restrict__ Dict,
              const float* __restrict__ cw,
              float* __restrict__ ws) {
  __shared__ float sh[Ksz];
  __shared__ float mx, sm;
  const int t = threadIdx.x;          // 64 threads, t == k
  float s = 0.f;
  const float* row = Dict + (size_t)t * Dsz;
  for (int j = 0; j < Dsz; ++j) s += row[j] * row[j];
  ws[t] = s;                          // d2[k]
  sh[t] = cw[t];
  __syncthreads();
  if (t == 0) {
    float m = sh[0];
    for (int k = 1; k < Ksz; ++k) m = fmaxf(m, sh[k]);
    float ss = 0.f;
    for (int k = 0; k < Ksz; ++k) ss += __expf(sh[k] - m);
    mx = m; sm = ss;
  }
  __syncthreads();
  ws[64 + t] = __expf(sh[t] - mx) / sm;  // norm_w[k]
}

__global__ __launch_bounds__(256)
void lde_main(const float* __restrict__ X,      // (L,B,D)
              const unsigned char* __restrict__ mask, // (B,L) bool
              const float* __restrict__ Dict,   // (K,D)
              const float* __restrict__ saw,    // (K,)
              const float* __restrict__ sab,    // (K,)
              const float* __restrict__ ws,     // d2 | nw
              float* __restrict__ S1p,          // (B*CHNK, D)
              float* __restrict__ Cp) {         // (B*CHNK, K)
  __shared__ float Xs[ROWS * XSTR];   // staged X rows
  __shared__ float Wd[ROWS * WSTR];   // xd tile, then softmax weights W
  __shared__ float aS[ROWS];          // per-row a_l = sum_k nw_k * W[l,k]

  const int t    = threadIdx.x;
  const int blk  = blockIdx.x;        // b*CHNK + chunk
  const int b    = blk / CHNK;
  const int ch   = blk % CHNK;
  const int l0   = ch * ROWS;

  const int wid  = t >> 5;            // wave 0..7
  const int lane = t & 31;
  const int h    = lane >> 4;         // half-wave
  const int ln   = lane & 15;
  const int wrow = wid >> 2;          // 0..1 (M tile)
  const int wcol = wid & 3;           // 0..3 (N tile)

  // ---- stage X[l0:l0+32, b, :] into LDS via async global->LDS B128 copies ----
  // (no VGPR round trip; tracked by ASYNCcnt; 16B-aligned src and dst)
  {
    #pragma unroll
    for (int i = 0; i < 8; ++i) {
      int idx = t + i * 256;          // 0..2047 float4 chunks
      int r   = idx >> 6;             // row 0..31
      int c4  = idx & 63;             // float4 column
      const float4* gsrc = (const float4*)X
          + (((size_t)(l0 + r) * Bsz + b) * (Dsz / 4) + c4);
      float* dst = &Xs[r * XSTR + c4 * 4];
      uint64_t ga = (uint64_t)(uintptr_t)gsrc;     // 64-bit global address
      uint32_t la = (uint32_t)(uintptr_t)dst;      // low 32 bits == LDS offset
      asm volatile("global_load_async_to_lds_b128 %0, %1, off"
                   :: "v"(la), "v"(ga) : "memory");
    }
  }

  // warm L2/WGP$ with this wave's Dict rows while the async copies fly
  const float* brow = Dict + (size_t)(wcol * 16 + ln) * Dsz;     // B: N=ln col (Dict row)
  __builtin_prefetch(brow, 0, 0);
  __builtin_prefetch(brow + 128, 0, 0);

  // wait for our async LDS writes, then barrier so all rows are visible
  asm volatile("s_wait_asynccnt 0x0" ::: "memory");
  __syncthreads();

  // ---- WMMA: xd = X_chunk (32xD) * Dict^T (DxK), 2x4 grid of 16x16 tiles ----
  v8f acc = {0.f,0.f,0.f,0.f,0.f,0.f,0.f,0.f};
  const float* arow = &Xs[(wrow * 16 + ln) * XSTR];              // A: M=ln row
  #pragma unroll 4
  for (int kb = 0; kb < Dsz; kb += 4) {
    // ISA 7.12.2: 32-bit A 16x4 -> lane(h,ln): v0=K(2h), v1=K(2h+1); B 4x16 analogous.
    v2f av = *(const v2f*)(arow + kb + 2 * h);
    v2f bv = *(const v2f*)(brow + kb + 2 * h);
    acc = __builtin_amdgcn_wmma_f32_16x16x4_f32(false, av, false, bv,
                                                (short)0, acc, false, false);
  }
  // C/D layout: VGPR i -> M = i + 8*h, N = ln
  #pragma unroll
  for (int i = 0; i < 8; ++i) {
    int r = wrow * 16 + i + 8 * h;
    Wd[r * WSTR + wcol * 16 + ln] = acc[i];
  }
  __syncthreads();

  // ---- per-row masked softmax over K=64 (wave handles 4 rows; lane -> k, k+32) ----
  const float d20 = ws[lane],        d21 = ws[lane + 32];
  const float nw0 = ws[64 + lane],   nw1 = ws[64 + lane + 32];
  const float w0  = saw[lane],       w1  = saw[lane + 32];
  const float bb0 = sab[lane],       bb1 = sab[lane + 32];

  for (int rr = 0; rr < 4; ++rr) {
    int r = wid * 4 + rr;
    // x2 = ||x_row||^2 via wave reduction
    float x2 = 0.f;
    const float* xr = &Xs[r * XSTR];
    #pragma unroll
    for (int j = 0; j < 8; ++j) { float v = xr[lane + j * 32]; x2 += v * v; }
    #pragma unroll
    for (int m = 16; m >= 1; m >>= 1) x2 += __shfl_xor(x2, m, 32);

    int msk = mask[(size_t)b * Lsz + l0 + r];
    float xd0 = Wd[r * WSTR + lane];
    float xd1 = Wd[r * WSTR + lane + 32];
    float dist0 = msk ? 10000.f : (x2 - 2.f * xd0 + d20);
    float dist1 = msk ? 10000.f : (x2 - 2.f * xd1 + d21);
    float lg0 = -w0 * dist0 + bb0;
    float lg1 = -w1 * dist1 + bb1;
    float mx = fmaxf(lg0, lg1);
    #pragma unroll
    for (int m = 16; m >= 1; m >>= 1) mx = fmaxf(mx, __shfl_xor(mx, m, 32));
    float e0 = __expf(lg0 - mx);
    float e1 = __expf(lg1 - mx);
    float s = e0 + e1;
    #pragma unroll
    for (int m = 16; m >= 1; m >>= 1) s += __shfl_xor(s, m, 32);
    float inv = 1.f / s;
    float W0 = e0 * inv, W1 = e1 * inv;
    Wd[r * WSTR + lane]      = W0;   // overwrite xd with W (rows private per wave)
    Wd[r * WSTR + lane + 32] = W1;
    float ar = nw0 * W0 + nw1 * W1;
    #pragma unroll
    for (int m = 16; m >= 1; m >>= 1) ar += __shfl_xor(ar, m, 32);
    if (lane == 0) aS[r] = ar;
  }
  __syncthreads();

  // ---- deterministic per-chunk partials (no atomics) ----
  if (t < Ksz) {                        // Cp[k] = sum_l W[l,k]
    float c = 0.f;
    for (int r = 0; r < ROWS; ++r) c += Wd[r * WSTR + t];
    Cp[(size_t)blk * Ksz + t] = c;
  }
  {                                     // S1p[d] = sum_l a_l * X[l,d]
    float s1 = 0.f;
    for (int r = 0; r < ROWS; ++r) s1 += aS[r] * Xs[r * XSTR + t];
    S1p[(size_t)blk * Dsz + t] = s1;
  }
}

__global__ __launch_bounds__(256)
void lde_final(const float* __restrict__ Dict,
               const float* __restrict__ ws,
               const float* __restrict__ S1p,
               const float* __restrict__ Cp,
               float* __restrict__ out) {
  __shared__ float cb[Ksz];
  const int b = blockIdx.x;
  const int t = threadIdx.x;            // t == d
  if (t < Ksz) {
    float c = 0.f;
    for (int chn = 0; chn < CHNK; ++chn) c += Cp[((size_t)b * CHNK + chn) * Ksz + t];
    cb[t] = ws[64 + t] * c;             // nw_k * C_k
  }
  __syncthreads();
  float s1 = 0.f;
  for (int chn = 0; chn < CHNK; ++chn) s1 += S1p[((size_t)b * CHNK + chn) * Dsz + t];
  float accd = 0.f;
  for (int k = 0; k < Ksz; ++k) accd += cb[k] * Dict[(size_t)k * Dsz + t];
  out[(size_t)b * Dsz + t] = (s1 - accd) * (1.0f / (float)Lsz);
}

extern "C" void kernel_launch(void* const* d_in, const int* in_sizes, int n_in,
                              void* d_out, int out_size, void* d_ws, size_t ws_size,
                              hipStream_t stream) {
  const float* X             = (const float*)d_in[0];           // (L,B,D)
  const unsigned char* mask  = (const unsigned char*)d_in[1];   // (B,L) bool (1 byte/elem)
  const float* Dict          = (const float*)d_in[2];           // (K,D)
  const float* cw            = (const float*)d_in[3];           // (K,)
  const float* saw           = (const float*)d_in[4];           // (K,)
  const float* sab           = (const float*)d_in[5];           // (K,)

  float* ws  = (float*)d_ws;
  float* S1p = ws + 128;                              // B*CHNK*Dsz floats
  float* Cp  = S1p + (size_t)Bsz * CHNK * Dsz;        // B*CHNK*Ksz floats
  // total ws use: (128 + 8*32*256 + 8*32*64) * 4 B ~= 328 KB; fully rewritten
  // every launch (no zero-init dependence) -> graph-replay safe & deterministic.

  lde_prep <<<1,          64,  0, stream>>>(Dict, cw, ws);
  lde_main <<<Bsz * CHNK, 256, 0, stream>>>(X, mask, Dict, saw, sab, ws, S1p, Cp);
  lde_final<<<Bsz,        256, 0, stream>>>(Dict, ws, S1p, Cp, (float*)d_out);
}